// COMSAGPool_gumble_89060441850429
// MI455X (gfx1250) — compile-verified
//
#include <hip/hip_runtime.h>
#include <hip/hip_bf16.h>

// ---- problem constants (match reference) ----
#define NN 100000    // nodes
#define EE 1600000   // edges
#define DD 128       // feature dim

typedef __attribute__((ext_vector_type(2))) float v2f;
typedef __attribute__((ext_vector_type(8))) float v8f;
typedef __attribute__((ext_vector_type(4))) unsigned int v4u;
typedef __attribute__((ext_vector_type(8))) int v8i;
typedef __attribute__((ext_vector_type(4))) int v4i;

// ---------------- zero fill ----------------
__global__ void k_zero(float* __restrict__ p, size_t n) {
    size_t i = (size_t)blockIdx.x * blockDim.x + threadIdx.x;
    size_t st = (size_t)gridDim.x * blockDim.x;
    for (; i < n; i += st) p[i] = 0.0f;
}

// ---------------- degrees via float atomics ----------------
__global__ void k_degree(const int* __restrict__ src, const int* __restrict__ dst,
                         float* __restrict__ deg_out, float* __restrict__ deg_in) {
    int e = blockIdx.x * blockDim.x + threadIdx.x;
    if (e >= EE) return;
    atomicAdd(&deg_out[src[e]], 1.0f);
    atomicAdd(&deg_in[dst[e]], 1.0f);
}

// ---------------- deg -> rsqrt(clip(deg,1)) in place ----------------
__global__ void k_norm(float* __restrict__ ns, float* __restrict__ nd) {
    int i = blockIdx.x * blockDim.x + threadIdx.x;
    if (i >= NN) return;
    ns[i] = rsqrtf(fmaxf(ns[i], 1.0f));
    nd[i] = rsqrtf(fmaxf(nd[i], 1.0f));
}

// ---------------- SpMM #1: agg1[dst] += feature[src] * ns[src] ----------------
// one wave32 per edge; each lane handles 4 consecutive floats (float4)
__global__ void k_spmm1(const float* __restrict__ feat, const int* __restrict__ src,
                        const int* __restrict__ dst, const float* __restrict__ ns,
                        float* __restrict__ agg1) {
    int e = blockIdx.x * 8 + (threadIdx.x >> 5);
    if (e >= EE) return;
    int lane = threadIdx.x & 31;
    int s = src[e], d = dst[e];
    float w = ns[s];
    const float4* fr = reinterpret_cast<const float4*>(feat + (size_t)s * DD);
    float4 v = fr[lane];
    float* out = agg1 + (size_t)d * DD + lane * 4;
    atomicAdd(out + 0, v.x * w);
    atomicAdd(out + 1, v.y * w);
    atomicAdd(out + 2, v.z * w);
    atomicAdd(out + 3, v.w * w);
}

// ---------------- GEMM1: out1 = (agg1 * nd[:,None]) @ W1 + b1 ----------------
// fp32 WMMA 16x16x4. Block = 256 thr = 8 waves; block owns 16 rows x 128 cols,
// wave w owns the 16x16 tile at columns [16w, 16w+16).
// W1 is staged into LDS by ONE tensor_load_to_lds DMA (TDM), using the TDM pad
// feature (1 dword of pad after every 128 dwords) to produce the 129-float
// bank-conflict-free LDS row stride in hardware. A stripe staged by VALU (needs
// the nd row-scale fused in).
#define WS_STRIDE 129   // = 128 data + 1 TDM pad dword per row
#define AS_STRIDE 132   // pad 128->132 so row-major column reads spread banks
__global__ void k_gemm1(const float* __restrict__ agg1, const float* __restrict__ nd,
                        const float* __restrict__ W1, const float* __restrict__ b1,
                        float* __restrict__ out1) {
    __shared__ float Ws[DD * WS_STRIDE];
    __shared__ float As[16 * AS_STRIDE];
    const int m0 = blockIdx.x * 16;
    const int tid = threadIdx.x;

    // ---- stage W1 (128x128) into LDS via Tensor Data Mover (wave 0 only) ----
    if (tid < 32) {
        const unsigned long long gaddr = (unsigned long long)(uintptr_t)W1;
        const unsigned int lds_off = (unsigned int)(uintptr_t)(void*)Ws; // low 32b = LDS byte offset

        // D# group 0: count=1 | lds_addr | global_addr | type=2
        v4u g0;
        g0.x = 1u;                                  // count=1, no gather, user desc
        g0.y = lds_off;                             // lds_addr (bytes)
        g0.z = (unsigned int)gaddr;                 // global_addr[31:0]
        g0.w = (unsigned int)((gaddr >> 32) & 0x01FFFFFFull) | (0x2u << 30); // [56:32] | type=2

        // D# group 1: data_size=4B | pad_enable, interval=128 dw, amount=1 dw
        //             tensor 128x128, tile 128x128, dim0_stride=128
        v8i g1;
        g1[0] = (int)((2u << 16) | (1u << 20) | (6u << 22)); // data_size=2(4B), pad_en, interval code 6
        g1[1] = (int)(128u << 16);                  // tensor_dim0 = 128 (low 16 in [63:48])
        g1[2] = (int)(128u << 16);                  // dim0 hi=0, tensor_dim1 = 128 (low 16)
        g1[3] = (int)(128u << 16);                  // dim1 hi=0, tile_dim0 = 128
        g1[4] = 128;                                // tile_dim1 = 128, tile_dim2 = 0
        g1[5] = 128;                                // tensor_dim0_stride = 128 (low 32)
        g1[6] = 0;                                  // stride hi, dim1_stride lo
        g1[7] = 0;

        v4i g2 = {};                                // 2D tensor: groups 2/3 unused
        v4i g3 = {};
        v8i g4 = {};                                // extra group (clang-23 6-arg form)
        __builtin_amdgcn_tensor_load_to_lds(g0, g1, g2, g3, g4, 0);
    }

    // ---- stage 16-row A stripe, pre-scaled by dest norm nd (fused) ----
    for (int i = tid; i < 16 * DD; i += 256) {
        int r = i >> 7, c = i & 127;
        As[r * AS_STRIDE + c] = agg1[(size_t)(m0 + r) * DD + c] * nd[m0 + r];
    }

    if (tid < 32) {
        __builtin_amdgcn_s_wait_tensorcnt(0);       // wave 0: DMA complete
    }
    __syncthreads();                                // publish LDS to all waves

    const int w = tid >> 5;
    const int lane = tid & 31;
    const int hf = lane >> 4;       // 0: K pair {k,k+1}, 1: {k+2,k+3}
    const int l = lane & 15;
    const int n0 = w * 16;

    v8f acc = {};
#pragma unroll
    for (int k = 0; k < DD; k += 4) {
        const int kk = k + 2 * hf;
        v2f a, b;
        // A (16x4, MxK): lanes 0-15 carry M=l with K={k,k+1}; lanes 16-31 K={k+2,k+3}
        a.x = As[l * AS_STRIDE + kk];
        a.y = As[l * AS_STRIDE + kk + 1];
        // B (4x16, KxN): lanes 0-15 carry N=l with K={k,k+1}; lanes 16-31 K={k+2,k+3}
        b.x = Ws[kk * WS_STRIDE + n0 + l];
        b.y = Ws[(kk + 1) * WS_STRIDE + n0 + l];
        acc = __builtin_amdgcn_wmma_f32_16x16x4_f32(
            /*neg_a=*/false, a, /*neg_b=*/false, b,
            /*c_mod=*/(short)0, acc, /*reuse_a=*/false, /*reuse_b=*/false);
    }

    // C/D layout: VGPR i -> M = i (lanes 0-15) or M = i+8 (lanes 16-31), N = lane%16
    const int col = n0 + l;
    const float bias = b1[col];
#pragma unroll
    for (int i = 0; i < 8; ++i) {
        const int row = m0 + i + 8 * hf;
        out1[(size_t)row * DD + col] = acc[i] + bias;
    }
}

// ---------------- GEMM2: h2 = out1 @ W2  (128 -> 2) ----------------
__global__ void k_gemm2(const float* __restrict__ out1, const float* __restrict__ W2,
                        float* __restrict__ h2) {
    __shared__ float W2s[DD * 2];
    int tid = threadIdx.x;
    if (tid < DD * 2) W2s[tid] = W2[tid];
    __syncthreads();
    int n = blockIdx.x * blockDim.x + tid;
    if (n >= NN) return;
    const float* row = out1 + (size_t)n * DD;
    float a0 = 0.0f, a1 = 0.0f;
#pragma unroll 4
    for (int k = 0; k < DD; ++k) {
        float x = row[k];
        a0 += x * W2s[2 * k];
        a1 += x * W2s[2 * k + 1];
    }
    h2[2 * n] = a0;
    h2[2 * n + 1] = a1;
}

// ---------------- SpMM #2 on the 2-wide result ----------------
__global__ void k_spmm2(const float* __restrict__ h2, const int* __restrict__ src,
                        const int* __restrict__ dst, const float* __restrict__ ns,
                        float* __restrict__ agg2) {
    int e = blockIdx.x * blockDim.x + threadIdx.x;
    if (e >= EE) return;
    int s = src[e], d = dst[e];
    float w = ns[s];
    atomicAdd(&agg2[2 * d + 0], h2[2 * s + 0] * w);
    atomicAdd(&agg2[2 * d + 1], h2[2 * s + 1] * w);
}

// ---------------- epilogue: gumbel hard mask + softmax/tanh gating ----------------
__global__ void k_final(const float* __restrict__ feat, const float* __restrict__ agg2,
                        const float* __restrict__ nd, const float* __restrict__ b2,
                        const float* __restrict__ gumbel, float* __restrict__ dout) {
    size_t i = (size_t)blockIdx.x * blockDim.x + threadIdx.x;
    if (i >= (size_t)NN * DD) return;
    int n = (int)(i >> 7);
    int c = (int)(i & 127);

    float sc = nd[n];
    float o0 = agg2[2 * n + 0] * sc + b2[0];
    float o1 = agg2[2 * n + 1] * sc + b2[1];

    // hard gumbel argmax (jnp.argmax picks first index on ties -> >=)
    float g0 = o0 + gumbel[2 * n + 0];
    float g1 = o1 + gumbel[2 * n + 1];
    float m0 = (g0 >= g1) ? 1.0f : 0.0f;
    float m1 = 1.0f - m0;

    // softmax(out2)
    float mx = fmaxf(o0, o1);
    float e0 = __expf(o0 - mx);
    float e1 = __expf(o1 - mx);
    float inv = 1.0f / (e0 + e1);
    float t0 = tanhf(e0 * inv) * m0;
    float t1 = tanhf(e1 * inv) * m1;

    float f = feat[i];
    float* dis = dout;
    float* com = dout + (size_t)NN * DD;
    dis[i] = f * t0;
    com[i] = f * t1;
    if (c == 0) {
        float* mk0 = dout + 2 * (size_t)NN * DD;
        float* mk1 = mk0 + NN;
        mk0[n] = m0;
        mk1[n] = m1;
    }
}

extern "C" void kernel_launch(void* const* d_in, const int* in_sizes, int n_in,
                              void* d_out, int out_size, void* d_ws, size_t ws_size,
                              hipStream_t stream) {
    const float* feature = (const float*)d_in[0];
    const int*   src     = (const int*)d_in[1];
    const int*   dst     = (const int*)d_in[2];
    const float* gumbel  = (const float*)d_in[3];
    // d_in[4] = label (unused by the forward outputs)
    const float* W1      = (const float*)d_in[5];
    const float* b1      = (const float*)d_in[6];
    const float* W2      = (const float*)d_in[7];
    const float* b2      = (const float*)d_in[8];
    float* out = (float*)d_out;

    // workspace layout (floats): ns | nd | agg2 | agg1 | out1 | h2
    float* ws    = (float*)d_ws;
    float* ns    = ws;                                 // N
    float* nd    = ws + NN;                            // N
    float* agg2  = ws + 2 * (size_t)NN;                // 2N
    float* agg1  = ws + 4 * (size_t)NN;                // N*D
    float* out1  = agg1 + (size_t)NN * DD;             // N*D
    float* h2    = out1 + (size_t)NN * DD;             // 2N

    // zero the accumulated prefix: ns, nd, agg2, agg1 (contiguous)
    size_t nzero = 4 * (size_t)NN + (size_t)NN * DD;
    k_zero<<<2048, 256, 0, stream>>>(ws, nzero);

    // degrees -> norms
    k_degree<<<(EE + 255) / 256, 256, 0, stream>>>(src, dst, ns, nd);
    k_norm<<<(NN + 255) / 256, 256, 0, stream>>>(ns, nd);

    // SpMM #1 (wave per edge)
    k_spmm1<<<(EE + 7) / 8, 256, 0, stream>>>(feature, src, dst, ns, agg1);

    // GEMM1 with fp32 WMMA + TDM-staged W1 (16 rows per block, 100000 = 6250*16)
    k_gemm1<<<NN / 16, 256, 0, stream>>>(agg1, nd, W1, b1, out1);

    // GEMM2 (128 -> 2)
    k_gemm2<<<(NN + 255) / 256, 256, 0, stream>>>(out1, W2, h2);

    // SpMM #2
    k_spmm2<<<(EE + 255) / 256, 256, 0, stream>>>(h2, src, dst, ns, agg2);

    // epilogue
    size_t tot = (size_t)NN * DD;
    k_final<<<(unsigned)((tot + 255) / 256), 256, 0, stream>>>(feature, agg2, nd, b2, gumbel, out);
}